// DINAttentionLayer_65429531787431
// MI455X (gfx1250) — compile-verified
//
#include <hip/hip_runtime.h>
#include <hip/hip_bf16.h>

typedef float v2f __attribute__((ext_vector_type(2)));
typedef float v8f __attribute__((ext_vector_type(8)));

#define B_   1024
#define T_   200
#define D_   64
#define M_   (B_ * T_)   // 204800, divisible by 128
#define K0_  256
#define H0_  80
#define H1_  40
#define EPS_ 1e-9f

// ---------------------------------------------------------------------------
// Kernel 0: zero the stats accumulators in workspace
// ---------------------------------------------------------------------------
__global__ void zero_kernel(float* __restrict__ p, int n) {
    int i = blockIdx.x * blockDim.x + threadIdx.x;
    if (i < n) p[i] = 0.f;
}

// ---------------------------------------------------------------------------
// Kernel 1: h0pre = concat(q,k,q-k,q*k) @ W0 + b0 ; fused sum/sumsq per feature
// 1 wave computes a 32-row M-tile x 80 cols (two 16-row A groups share every
// B fragment -> 1 global B load per 2 wmma instead of 1).
// ---------------------------------------------------------------------------
__global__ __launch_bounds__(64) void gemm0_kernel(
    const float* __restrict__ q, const float* __restrict__ key,
    const float* __restrict__ W0, const float* __restrict__ b0,
    float* __restrict__ h0pre, float* __restrict__ gsum, float* __restrict__ gsq)
{
    __shared__ __align__(16) float ks[2][32 * 68];   // pad 68: conflict-free, 16B aligned
    __shared__ __align__(16) float qs[2][32 * 68];
    __shared__ float ssum[H0_], ssq[H0_];

    const int tid  = threadIdx.x;
    const int lane = tid & 31;
    const int w    = tid >> 5;

    for (int i = tid; i < H0_; i += 64) { ssum[i] = 0.f; ssq[i] = 0.f; }

    const int tile = blockIdx.x * 2 + w;   // 32-row tiles
    const int row0 = tile * 32;

    // Coalesced staging of this wave's 32 rows of key and (row-matched) query.
    for (int j = 0; j < 16; ++j) {
        int idx = j * 32 + lane;          // 0..511
        int r   = idx >> 4;               // row in tile (0..31)
        int c4  = idx & 15;               // float4 column
        int m   = row0 + r;
        int b   = m / T_;
        int t   = m - b * T_;
        float4 kv = *(const float4*)(key + ((size_t)b * T_ + t) * D_ + c4 * 4);
        float4 qv = *(const float4*)(q   + (size_t)b * D_ + c4 * 4);
        *(float4*)(&ks[w][r * 68 + c4 * 4]) = kv;
        *(float4*)(&qs[w][r * 68 + c4 * 4]) = qv;
    }
    __syncthreads();

    v8f acc[2][5];
    for (int g = 0; g < 2; ++g)
        for (int nt = 0; nt < 5; ++nt)
            for (int i = 0; i < 8; ++i) acc[g][nt][i] = 0.f;

    const int r  = lane & 15;   // A row within group / D column-within-tile
    const int hi = lane >> 4;
    const float* qs0 = &qs[w][r * 68];
    const float* ks0 = &ks[w][r * 68];
    const float* qs1 = &qs[w][(r + 16) * 68];
    const float* ks1 = &ks[w][(r + 16) * 68];

    for (int s = 0; s < 4; ++s) {                 // concat section
        for (int kk = 0; kk < 16; ++kk) {         // 16 K-steps of 4 per section
            int c0 = kk * 4 + hi * 2;
            float qa0 = qs0[c0], qa1 = qs0[c0 + 1];
            float ka0 = ks0[c0], ka1 = ks0[c0 + 1];
            float qb0 = qs1[c0], qb1 = qs1[c0 + 1];
            float kb0 = ks1[c0], kb1 = ks1[c0 + 1];
            v2f a0, a1;
            if (s == 0)      { a0.x = qa0;       a0.y = qa1;       a1.x = qb0;       a1.y = qb1;       }
            else if (s == 1) { a0.x = ka0;       a0.y = ka1;       a1.x = kb0;       a1.y = kb1;       }
            else if (s == 2) { a0.x = qa0 - ka0; a0.y = qa1 - ka1; a1.x = qb0 - kb0; a1.y = qb1 - kb1; }
            else             { a0.x = qa0 * ka0; a0.y = qa1 * ka1; a1.x = qb0 * kb0; a1.y = qb1 * kb1; }

            int kg = s * 64 + kk * 4 + hi * 2;
            const float* w0p = W0 + (size_t)kg * H0_ + r;   // n = nt*16 + r
            for (int nt = 0; nt < 5; ++nt) {
                v2f bf;
                bf.x = w0p[nt * 16];
                bf.y = w0p[H0_ + nt * 16];
                acc[0][nt] = __builtin_amdgcn_wmma_f32_16x16x4_f32(
                    false, a0, false, bf, (short)0, acc[0][nt], false, false);
                acc[1][nt] = __builtin_amdgcn_wmma_f32_16x16x4_f32(
                    false, a1, false, bf, (short)0, acc[1][nt], false, false);
            }
        }
    }

    // bias + store + per-feature partial stats
    for (int nt = 0; nt < 5; ++nt) {
        int n = nt * 16 + r;
        float bias = b0[n];
        float ps = 0.f, pq = 0.f;
        for (int g = 0; g < 2; ++g) {
            for (int rr = 0; rr < 8; ++rr) {
                int m = g * 16 + rr + hi * 8;
                float v = acc[g][nt][rr] + bias;
                h0pre[(size_t)(row0 + m) * H0_ + n] = v;
                ps += v; pq += v * v;
            }
        }
        atomicAdd(&ssum[n], ps);
        atomicAdd(&ssq[n], pq);
    }
    __syncthreads();
    for (int i = tid; i < H0_; i += 64) {
        atomicAdd(&gsum[i], ssum[i]);
        atomicAdd(&gsq[i],  ssq[i]);
    }
}

// ---------------------------------------------------------------------------
// Kernel 2/4: finalize mean & rsqrt(var + eps)
// ---------------------------------------------------------------------------
__global__ void finalize_kernel(const float* __restrict__ gsum, const float* __restrict__ gsq,
                                float* __restrict__ meanv, float* __restrict__ scalev,
                                int n, float invM)
{
    int i = blockIdx.x * blockDim.x + threadIdx.x;
    if (i < n) {
        float mu  = gsum[i] * invM;
        float var = gsq[i] * invM - mu * mu;
        meanv[i]  = mu;
        scalev[i] = rsqrtf(var + EPS_);
    }
}

// ---------------------------------------------------------------------------
// Kernel 3: h1pre = dice(h0pre) @ W1 + b1 ; fused stats. 32-row tiles per wave.
// ---------------------------------------------------------------------------
__global__ __launch_bounds__(128) void gemm1_kernel(
    const float* __restrict__ h0pre,
    const float* __restrict__ mean0, const float* __restrict__ scale0,
    const float* __restrict__ alpha0,
    const float* __restrict__ W1, const float* __restrict__ b1,
    float* __restrict__ h1pre, float* __restrict__ gsum, float* __restrict__ gsq)
{
    __shared__ float sm[H0_], ss[H0_], sa[H0_];
    __shared__ float ssum[H1_], ssq[H1_];

    const int tid  = threadIdx.x;
    const int lane = tid & 31;
    const int w    = tid >> 5;

    for (int i = tid; i < H0_; i += 128) { sm[i] = mean0[i]; ss[i] = scale0[i]; sa[i] = alpha0[i]; }
    for (int i = tid; i < H1_; i += 128) { ssum[i] = 0.f; ssq[i] = 0.f; }
    __syncthreads();

    const int tile = blockIdx.x * 4 + w;   // 32-row tiles
    const int row0 = tile * 32;
    const int r  = lane & 15;
    const int hi = lane >> 4;

    v8f acc[2][3];
    for (int g = 0; g < 2; ++g)
        for (int nt = 0; nt < 3; ++nt)
            for (int i = 0; i < 8; ++i) acc[g][nt][i] = 0.f;

    const float* arow0 = h0pre + (size_t)(row0 + r) * H0_;
    const float* arow1 = h0pre + (size_t)(row0 + 16 + r) * H0_;

    for (int kk = 0; kk < 20; ++kk) {          // K = 80 -> 20 steps of 4
        int k0 = kk * 4 + hi * 2;
        float mu0 = sm[k0], mu1 = sm[k0 + 1];
        float sc0 = ss[k0], sc1 = ss[k0 + 1];
        float al0 = sa[k0], al1 = sa[k0 + 1];

        float xa0 = arow0[k0], xa1 = arow0[k0 + 1];
        float xb0 = arow1[k0], xb1 = arow1[k0 + 1];

        float pa0 = 1.f / (1.f + __expf(-(xa0 - mu0) * sc0));
        float pa1 = 1.f / (1.f + __expf(-(xa1 - mu1) * sc1));
        float pb0 = 1.f / (1.f + __expf(-(xb0 - mu0) * sc0));
        float pb1 = 1.f / (1.f + __expf(-(xb1 - mu1) * sc1));

        v2f a0, a1;
        a0.x = xa0 * (pa0 + al0 * (1.f - pa0));
        a0.y = xa1 * (pa1 + al1 * (1.f - pa1));
        a1.x = xb0 * (pb0 + al0 * (1.f - pb0));
        a1.y = xb1 * (pb1 + al1 * (1.f - pb1));

        for (int nt = 0; nt < 3; ++nt) {
            int n = nt * 16 + r;
            v2f bf;
            bf.x = (n < H1_) ? W1[(size_t)k0 * H1_ + n]       : 0.f;
            bf.y = (n < H1_) ? W1[(size_t)(k0 + 1) * H1_ + n] : 0.f;
            acc[0][nt] = __builtin_amdgcn_wmma_f32_16x16x4_f32(
                false, a0, false, bf, (short)0, acc[0][nt], false, false);
            acc[1][nt] = __builtin_amdgcn_wmma_f32_16x16x4_f32(
                false, a1, false, bf, (short)0, acc[1][nt], false, false);
        }
    }

    for (int nt = 0; nt < 3; ++nt) {
        int n = nt * 16 + r;
        if (n < H1_) {
            float bias = b1[n];
            float ps = 0.f, pq = 0.f;
            for (int g = 0; g < 2; ++g) {
                for (int rr = 0; rr < 8; ++rr) {
                    int m = g * 16 + rr + hi * 8;
                    float v = acc[g][nt][rr] + bias;
                    h1pre[(size_t)(row0 + m) * H1_ + n] = v;
                    ps += v; pq += v * v;
                }
            }
            atomicAdd(&ssum[n], ps);
            atomicAdd(&ssq[n], pq);
        }
    }
    __syncthreads();
    if (tid < H1_) {
        atomicAdd(&gsum[tid], ssum[tid]);
        atomicAdd(&gsq[tid],  ssq[tid]);
    }
}

// ---------------------------------------------------------------------------
// Kernel 5: dice1 + Wout projection + masked softmax + attn @ key (1 block / b)
// ---------------------------------------------------------------------------
__global__ __launch_bounds__(256) void attn_kernel(
    const float* __restrict__ h1pre, const float* __restrict__ key,
    const unsigned char* __restrict__ mask,
    const float* __restrict__ mean1, const float* __restrict__ scale1,
    const float* __restrict__ alpha1,
    const float* __restrict__ Wout, const float* __restrict__ bout,
    float* __restrict__ out)
{
    __shared__ float sc[T_];
    __shared__ float sw[H1_], sm[H1_], ssc[H1_], sa[H1_];
    __shared__ float red[256];
    __shared__ float part[4][D_];

    const int tid = threadIdx.x;
    const int b   = blockIdx.x;

    if (tid < H1_) { sw[tid] = Wout[tid]; sm[tid] = mean1[tid]; ssc[tid] = scale1[tid]; sa[tid] = alpha1[tid]; }
    __syncthreads();

    if (tid < T_) {
        const float* hrow = h1pre + ((size_t)b * T_ + tid) * H1_;
        float s = bout[0];
        for (int j = 0; j < H1_; ++j) {
            float x = hrow[j];
            float z = (x - sm[j]) * ssc[j];
            float p = 1.f / (1.f + __expf(-z));
            s += x * (p + sa[j] * (1.f - p)) * sw[j];
        }
        bool mk = mask[(size_t)b * T_ + tid] != 0;
        sc[tid] = mk ? s : -2.0e32f;     // (-2e32 + 1) rounds to -2e32 in fp32
    }
    __syncthreads();

    // max
    float v = (tid < T_) ? sc[tid] : -3.0e38f;
    red[tid] = v; __syncthreads();
    for (int off = 128; off > 0; off >>= 1) {
        if (tid < off) red[tid] = fmaxf(red[tid], red[tid + off]);
        __syncthreads();
    }
    float mx = red[0];
    __syncthreads();

    // exp + sum
    float e = (tid < T_) ? __expf(sc[tid] - mx) : 0.f;
    red[tid] = e; __syncthreads();
    for (int off = 128; off > 0; off >>= 1) {
        if (tid < off) red[tid] += red[tid + off];
        __syncthreads();
    }
    float inv = 1.f / red[0];
    __syncthreads();
    if (tid < T_) sc[tid] = e * inv;
    __syncthreads();

    // out[b,d] = sum_t attn[t] * key[b,t,d]  (coalesced over d; 4 t-chunks)
    const int d  = tid & 63;
    const int ch = tid >> 6;
    float accv = 0.f;
    for (int t = ch * 50; t < ch * 50 + 50; ++t)
        accv += sc[t] * key[((size_t)b * T_ + t) * D_ + d];
    part[ch][d] = accv;
    __syncthreads();
    if (tid < D_)
        out[(size_t)b * D_ + tid] = part[0][tid] + part[1][tid] + part[2][tid] + part[3][tid];
}

// ---------------------------------------------------------------------------
extern "C" void kernel_launch(void* const* d_in, const int* in_sizes, int n_in,
                              void* d_out, int out_size, void* d_ws, size_t ws_size,
                              hipStream_t stream)
{
    const float*         q      = (const float*)d_in[0];
    const float*         key    = (const float*)d_in[1];
    const unsigned char* mask   = (const unsigned char*)d_in[2];   // jax bool = 1 byte
    const float*         W0     = (const float*)d_in[3];
    const float*         b0     = (const float*)d_in[4];
    const float*         alpha0 = (const float*)d_in[5];
    const float*         W1     = (const float*)d_in[6];
    const float*         b1     = (const float*)d_in[7];
    const float*         alpha1 = (const float*)d_in[8];
    const float*         Wout   = (const float*)d_in[9];
    const float*         bout   = (const float*)d_in[10];

    float* ws     = (float*)d_ws;
    float* h0pre  = ws;                                   // M*80 floats (65.5 MB)
    float* h1pre  = h0pre + (size_t)M_ * H0_;             // M*40 floats (32.8 MB)
    float* stats  = h1pre + (size_t)M_ * H1_;             // 480 floats
    float* gsum0  = stats;        float* gsq0   = stats + 80;
    float* gsum1  = stats + 160;  float* gsq1   = stats + 200;
    float* mean0  = stats + 240;  float* scale0 = stats + 320;
    float* mean1  = stats + 400;  float* scale1 = stats + 440;

    const float invM = 1.f / (float)M_;

    zero_kernel<<<1, 512, 0, stream>>>(stats, 480);
    gemm0_kernel<<<M_ / 64, 64, 0, stream>>>(q, key, W0, b0, h0pre, gsum0, gsq0);
    finalize_kernel<<<1, 128, 0, stream>>>(gsum0, gsq0, mean0, scale0, H0_, invM);
    gemm1_kernel<<<M_ / 128, 128, 0, stream>>>(h0pre, mean0, scale0, alpha0, W1, b1,
                                               h1pre, gsum1, gsq1);
    finalize_kernel<<<1, 64, 0, stream>>>(gsum1, gsq1, mean1, scale1, H1_, invM);
    attn_kernel<<<B_, 256, 0, stream>>>(h1pre, key, mask, mean1, scale1, alpha1,
                                        Wout, bout, (float*)d_out);
}